// ECELoss_5282809774396
// MI455X (gfx1250) — compile-verified
//
#include <hip/hip_runtime.h>
#include <hip/hip_bf16.h>

#define N_BINS 15
#define WAVE 32                  // gfx1250 is wave32-only
#define BLOCK_THREADS 128        // 4 waves per block (keeps static LDS small)
#define WAVES_PER_BLOCK (BLOCK_THREADS / WAVE)
#define SLOT_FLOATS 48           // 15 cnt + 15 conf + 15 acc, padded to 48

// Async staging geometry for the C==128 fast path:
// each wave stages 32 rows x 32 floats (a "quarter row") per buffer,
// LDS row stride padded to 36 floats (144B) to spread banks.
#define Q_FLOATS 32
#define LROW 36
#define QBUF_FLOATS (32 * LROW)  // 1152 floats = 4608 B per buffer

// ---------------------------------------------------------------------------
// Per-row helpers
// ---------------------------------------------------------------------------
__device__ __forceinline__ void accumulate_row(float conf, float acc,
                                               float cnt[N_BINS],
                                               float csum[N_BINS],
                                               float asum[N_BINS])
{
    // bin i covers (i/15, (i+1)/15]  ->  bin = ceil(conf*15) - 1, clamped
    int b = (int)ceilf(conf * (float)N_BINS) - 1;
    b = b < 0 ? 0 : (b > N_BINS - 1 ? N_BINS - 1 : b);
#pragma unroll
    for (int k = 0; k < N_BINS; ++k) {
        float m = (k == b) ? 1.0f : 0.0f;
        cnt[k]  += m;
        csum[k] += m * conf;
        asum[k] += m * acc;
    }
}

__device__ __forceinline__ void direct_row(const float* __restrict__ rowp, int C,
                                           float* best_out, int* bidx_out)
{
    float best = -1.0f;          // uniform(0,1) inputs: always > -1
    int   bidx = 0;
    if ((C & 3) == 0) {
        const float4* rp = (const float4*)rowp;
        for (int j = 0; j < (C >> 2); ++j) {
            float4 v = rp[j];
            if (v.x > best) { best = v.x; bidx = 4 * j + 0; }
            if (v.y > best) { best = v.y; bidx = 4 * j + 1; }
            if (v.z > best) { best = v.z; bidx = 4 * j + 2; }
            if (v.w > best) { best = v.w; bidx = 4 * j + 3; }
        }
    } else {
        for (int j = 0; j < C; ++j) {
            float v = rowp[j];
            if (v > best) { best = v; bidx = j; }
        }
    }
    *best_out = best;
    *bidx_out = bidx;
}

// Issue one quarter: 8 async b128 copies; each op moves 32 lanes x 16B.
// INST_OFFSET is added to BOTH the LDS and global addresses (ISA 08 §4.4),
// so one base register pair + immediate offsets covers the whole quarter.
__device__ __forceinline__ void issue_quarter_async(unsigned lds_base,
                                                    const char* gsrc)
{
#pragma unroll
    for (int j = 0; j < 8; ++j) {
        asm volatile("global_load_async_to_lds_b128 %0, %1, off offset:%2"
                     :: "v"(lds_base), "v"(gsrc), "i"(j * 16)
                     : "memory");
    }
}

// ---------------------------------------------------------------------------
// Kernel 1: streaming pass.
// Fast path (C==128): wave-cooperative async staging, rows rb..rb+31 per
// group (contiguous 16KB), quarters double-buffered through LDS, lane L
// consumes row rb+L from its padded LDS row. Tail rows + generic C use
// direct loads. Deterministic: fixed row order, fixed-order butterfly.
// ---------------------------------------------------------------------------
__global__ void __launch_bounds__(BLOCK_THREADS)
ece_partial_kernel(const float* __restrict__ sm,
                   const int*   __restrict__ labels,
                   float*       __restrict__ wave_partials,
                   int N, int C, int total_threads)
{
    __shared__ float stage[WAVES_PER_BLOCK][2][QBUF_FLOATS];  // 36,864 B

    const int tid          = blockIdx.x * BLOCK_THREADS + threadIdx.x;
    const int lane         = threadIdx.x & (WAVE - 1);
    const int wave_in_blk  = threadIdx.x >> 5;
    const int wgid         = tid >> 5;                 // global wave id
    const int total_waves  = total_threads >> 5;

    float cnt[N_BINS], csum[N_BINS], asum[N_BINS];
#pragma unroll
    for (int k = 0; k < N_BINS; ++k) { cnt[k] = 0.f; csum[k] = 0.f; asum[k] = 0.f; }

    if (C == 128) {
        float* buf0 = &stage[wave_in_blk][0][0];
        float* buf1 = &stage[wave_in_blk][1][0];
        // Lane's LDS destination base (byte offset, low 32 bits of the
        // generic pointer are the LDS offset).
        const unsigned lds0 = (unsigned)(uintptr_t)buf0 + (unsigned)(lane * (LROW * 4));
        const unsigned lds1 = (unsigned)(uintptr_t)buf1 + (unsigned)(lane * (LROW * 4));
        const int group_stride = total_waves * WAVE;   // rows per sweep

        int rb = wgid * WAVE;
        for (; rb + WAVE <= N; rb += group_stride) {   // wave-uniform: EXEC all-1s
            // lane L streams row rb+L (512B); group spans 16KB contiguous
            const char* gsrc = (const char*)sm + (size_t)(rb + lane) * 512;

            float best = -1.0f;
            int   bidx = 0;

            issue_quarter_async(lds0, gsrc);           // prologue: q0 -> buf0
#pragma unroll
            for (int q = 0; q < 4; ++q) {
                if (q < 3) {
                    // don't overwrite a buffer with DS reads still in flight
                    asm volatile("s_wait_dscnt 0x0" ::: "memory");
                    issue_quarter_async((q & 1) ? lds0 : lds1,
                                        gsrc + (q + 1) * 128);
                    asm volatile("s_wait_asynccnt 0x8" ::: "memory");
                } else {
                    asm volatile("s_wait_asynccnt 0x0" ::: "memory");
                }
                const float4* lrow =
                    (const float4*)(((q & 1) ? buf1 : buf0) + lane * LROW);
#pragma unroll
                for (int j = 0; j < 8; ++j) {
                    float4 v = lrow[j];
                    const int ib = q * Q_FLOATS + 4 * j;
                    if (v.x > best) { best = v.x; bidx = ib + 0; }
                    if (v.y > best) { best = v.y; bidx = ib + 1; }
                    if (v.z > best) { best = v.z; bidx = ib + 2; }
                    if (v.w > best) { best = v.w; bidx = ib + 3; }
                }
            }
            const float acc = (bidx == labels[rb + lane]) ? 1.0f : 0.0f;
            accumulate_row(best, acc, cnt, csum, asum);
        }

        // Tail: at most one partial group per wave -> direct loads (divergent OK)
        if (rb < N) {
            const int row = rb + lane;
            if (row < N) {
                __builtin_prefetch(sm + (size_t)row * 128, 0, 1); // global_prefetch_b8
                float best; int bidx;
                direct_row(sm + (size_t)row * 128, 128, &best, &bidx);
                const float acc = (bidx == labels[row]) ? 1.0f : 0.0f;
                accumulate_row(best, acc, cnt, csum, asum);
            }
        }
    } else {
        // Generic path: thread-per-row grid-stride with direct loads
        for (int row = tid; row < N; row += total_threads) {
            if (row + total_threads < N)
                __builtin_prefetch(sm + (size_t)(row + total_threads) * (size_t)C, 0, 1);
            float best; int bidx;
            direct_row(sm + (size_t)row * (size_t)C, C, &best, &bidx);
            const float acc = (bidx == labels[row]) ? 1.0f : 0.0f;
            accumulate_row(best, acc, cnt, csum, asum);
        }
    }

    // fixed-order wave32 butterfly reduction (deterministic)
#pragma unroll
    for (int off = 16; off >= 1; off >>= 1) {
#pragma unroll
        for (int k = 0; k < N_BINS; ++k) {
            cnt[k]  += __shfl_xor(cnt[k],  off, WAVE);
            csum[k] += __shfl_xor(csum[k], off, WAVE);
            asum[k] += __shfl_xor(asum[k], off, WAVE);
        }
    }

    if (lane == 0) {
        float* slot = wave_partials + (size_t)wgid * SLOT_FLOATS;
#pragma unroll
        for (int k = 0; k < N_BINS; ++k) {
            slot[k]              = cnt[k];
            slot[N_BINS + k]     = csum[k];
            slot[2 * N_BINS + k] = asum[k];
        }
    }
}

// ---------------------------------------------------------------------------
// Kernel 2: deterministic fixed-order reduction of per-wave partials + ECE.
// 240 threads = 15 bins x 16 chunks; double precision accumulation.
// ---------------------------------------------------------------------------
__global__ void __launch_bounds__(240)
ece_final_kernel(const float* __restrict__ wave_partials,
                 int num_waves, int N, float* __restrict__ out)
{
    __shared__ double s[3][16][N_BINS];

    const int t     = threadIdx.x;       // 0..239
    const int bin   = t % N_BINS;
    const int chunk = t / N_BINS;        // 0..15

    double c = 0.0, cf = 0.0, ac = 0.0;
    for (int w = chunk; w < num_waves; w += 16) {
        const float* p = wave_partials + (size_t)w * SLOT_FLOATS;
        c  += (double)p[bin];
        cf += (double)p[N_BINS + bin];
        ac += (double)p[2 * N_BINS + bin];
    }
    s[0][chunk][bin] = c;
    s[1][chunk][bin] = cf;
    s[2][chunk][bin] = ac;
    __syncthreads();

    if (chunk == 0) {
        double C_ = 0.0, CF = 0.0, AC = 0.0;
#pragma unroll
        for (int q = 0; q < 16; ++q) {
            C_ += s[0][q][bin];
            CF += s[1][q][bin];
            AC += s[2][q][bin];
        }
        s[0][0][bin] = C_;
        s[1][0][bin] = CF;
        s[2][0][bin] = AC;
    }
    __syncthreads();

    if (t == 0) {
        double ece = 0.0;
        for (int k = 0; k < N_BINS; ++k) {
            double cc = s[0][0][k];
            if (cc > 0.0) {
                double gap = fabs(s[1][0][k] / cc - s[2][0][k] / cc);
                ece += gap * (cc / (double)N);
            }
        }
        out[0] = (float)ece;
    }
}

// ---------------------------------------------------------------------------
extern "C" void kernel_launch(void* const* d_in, const int* in_sizes, int n_in,
                              void* d_out, int out_size, void* d_ws, size_t ws_size,
                              hipStream_t stream)
{
    const float* softmaxes = (const float*)d_in[0];
    const int*   labels    = (const int*)d_in[1];
    float*       out       = (float*)d_out;
    float*       partials  = (float*)d_ws;

    const int N = in_sizes[1];                 // 1,000,000 rows
    const int C = (N > 0) ? (in_sizes[0] / N)  // 128 classes
                          : 0;

    // 2048 blocks x 4 waves = 8192 waves saturates HBM; shrink if the
    // workspace can't hold the per-wave partials (48 floats per wave).
    int blocks = 2048;
    const size_t bytes_per_block = (size_t)WAVES_PER_BLOCK * SLOT_FLOATS * sizeof(float);
    if ((size_t)blocks * bytes_per_block > ws_size) {
        blocks = (int)(ws_size / bytes_per_block);
        if (blocks < 1) blocks = 1;
    }
    const int total_threads = blocks * BLOCK_THREADS;
    const int num_waves     = total_threads / WAVE;

    ece_partial_kernel<<<blocks, BLOCK_THREADS, 0, stream>>>(
        softmaxes, labels, partials, N, C, total_threads);

    ece_final_kernel<<<1, 240, 0, stream>>>(partials, num_waves, N, out);
}